// NWKernelRegression_31980326486771
// MI455X (gfx1250) — compile-verified
//
#include <hip/hip_runtime.h>
#include <hip/hip_bf16.h>

// Nadaraya-Watson kernel regression, N rows x K keys/values (f32).
// Bandwidth-bound: 2*N*K*4B = 268 MB streamed once @ 23.3 TB/s => ~11.5 us floor.
// Strategy: 1 workgroup (256 thr / 8 waves, wave32) per row; 4-buffer,
// prefetch-distance-2 async global->LDS pipeline (CDNA5
// GLOBAL_LOAD_ASYNC_TO_LDS_B128 + s_wait_asynccnt) with non-temporal hints
// (dataset > 192 MB L2, touched once), exp2-based Gaussian weights on VALU,
// wave32 shuffle reduction + small LDS combine.

#define THREADS     256
#define TILE_ELEMS  1024   // floats per tile = THREADS * 4
#define NBUF        4      // LDS buffers; prefetch distance 2

// LDS byte offset (wave-relative) of a pointer into a __shared__ object.
__device__ __forceinline__ unsigned lds_addr_u32(const void* p) {
  return (unsigned)(unsigned long long)
         (__attribute__((address_space(3))) const void*)p;
}

// CDNA5 async copy: 16 bytes per lane, global -> LDS, tracked by ASYNCcnt.
// Non-temporal: streamed data, never reused -> don't pollute WGP$/L2.
__device__ __forceinline__ void async_copy_b128_nt(unsigned lds_off, const float* gaddr) {
  asm volatile("global_load_async_to_lds_b128 %0, %1, off th:TH_LOAD_NT"
               :: "v"(lds_off), "v"(gaddr)
               : "memory");
}

__device__ __forceinline__ void wait_async_le4() {
  asm volatile("s_wait_asynccnt 0x4" ::: "memory");
}
__device__ __forceinline__ void wait_async_le2() {
  asm volatile("s_wait_asynccnt 0x2" ::: "memory");
}
__device__ __forceinline__ void wait_async_le0() {
  asm volatile("s_wait_asynccnt 0x0" ::: "memory");
}

__global__ __launch_bounds__(THREADS)
void nw_regression_kernel(const float* __restrict__ queries,
                          const float* __restrict__ keys,
                          const float* __restrict__ values,
                          const float* __restrict__ w,
                          float* __restrict__ out,
                          int N, int K) {
  __shared__ float kbuf[NBUF][TILE_ELEMS];
  __shared__ float vbuf[NBUF][TILE_ELEMS];
  __shared__ float redE[THREADS / 32];
  __shared__ float redEV[THREADS / 32];

  const int n = blockIdx.x;
  if (n >= N) return;

  const int tid = threadIdx.x;
  const float qn = queries[n];
  const float wv = w[0];
  // exp(-0.5*d^2) == exp2(c2 * d^2), c2 = -0.5*log2(e)
  const float c2 = -0.72134752044448170f;

  const size_t rowbase = (size_t)n * (size_t)K;
  const float* krow = keys   + rowbase;
  const float* vrow = values + rowbase;

  unsigned koff[NBUF], voff[NBUF];
#pragma unroll
  for (int b = 0; b < NBUF; ++b) {
    koff[b] = lds_addr_u32(&kbuf[b][tid * 4]);
    voff[b] = lds_addr_u32(&vbuf[b][tid * 4]);
  }

  const int ntiles = K / TILE_ELEMS;  // K=4096 -> 4 tiles

  // Prologue: prefetch tiles 0 and 1 (2 async ops per wave per tile).
  async_copy_b128_nt(koff[0], krow + tid * 4);
  async_copy_b128_nt(voff[0], vrow + tid * 4);
  if (ntiles > 1) {
    async_copy_b128_nt(koff[1], krow + TILE_ELEMS + tid * 4);
    async_copy_b128_nt(voff[1], vrow + TILE_ELEMS + tid * 4);
  }

  float e = 0.0f, ev = 0.0f;

  for (int t = 0; t < ntiles; ++t) {
    const int buf = t & (NBUF - 1);
    if (t + 2 < ntiles) {
      // Keep two tiles in flight; ASYNCcnt completes in order, so after the
      // issue there are <=6 outstanding and <=4 means tile t has landed.
      const int pb = (t + 2) & (NBUF - 1);
      async_copy_b128_nt(koff[pb], krow + (size_t)(t + 2) * TILE_ELEMS + tid * 4);
      async_copy_b128_nt(voff[pb], vrow + (size_t)(t + 2) * TILE_ELEMS + tid * 4);
      wait_async_le4();
    } else if (t + 1 < ntiles) {
      wait_async_le2();   // only tile t+1 still in flight
    } else {
      wait_async_le0();   // drain
    }

    // Each thread reads back exactly the 16B it staged (per-wave ASYNCcnt
    // ordering suffices; no cross-wave LDS dependency, no barrier needed).
    const float4 kk = *(const float4*)&kbuf[buf][tid * 4];
    const float4 vv = *(const float4*)&vbuf[buf][tid * 4];

    float d, p;
    d = (qn - kk.x) * wv; p = __builtin_amdgcn_exp2f(c2 * d * d); e += p; ev += p * vv.x;
    d = (qn - kk.y) * wv; p = __builtin_amdgcn_exp2f(c2 * d * d); e += p; ev += p * vv.y;
    d = (qn - kk.z) * wv; p = __builtin_amdgcn_exp2f(c2 * d * d); e += p; ev += p * vv.z;
    d = (qn - kk.w) * wv; p = __builtin_amdgcn_exp2f(c2 * d * d); e += p; ev += p * vv.w;
  }

  // Wave32 reduction.
  for (int off = 16; off > 0; off >>= 1) {
    e  += __shfl_xor(e,  off, 32);
    ev += __shfl_xor(ev, off, 32);
  }
  const int wave = tid >> 5;
  if ((tid & 31) == 0) { redE[wave] = e; redEV[wave] = ev; }
  __syncthreads();
  if (tid == 0) {
    float te = 0.0f, tev = 0.0f;
#pragma unroll
    for (int i = 0; i < THREADS / 32; ++i) { te += redE[i]; tev += redEV[i]; }
    out[n] = tev / te;
  }
}

extern "C" void kernel_launch(void* const* d_in, const int* in_sizes, int n_in,
                              void* d_out, int out_size, void* d_ws, size_t ws_size,
                              hipStream_t stream) {
  const float* queries = (const float*)d_in[0];
  const float* keys    = (const float*)d_in[1];
  const float* values  = (const float*)d_in[2];
  const float* w       = (const float*)d_in[3];
  float* out = (float*)d_out;

  const int N = in_sizes[0];
  const int K = in_sizes[1] / N;   // 4096; must be a multiple of TILE_ELEMS

  nw_regression_kernel<<<dim3(N), dim3(THREADS), 0, stream>>>(
      queries, keys, values, w, out, N, K);
}